// GPTUniAttention_87883620810956
// MI455X (gfx1250) — compile-verified
//
#include <hip/hip_runtime.h>

typedef __attribute__((ext_vector_type(2))) float v2f;
typedef __attribute__((ext_vector_type(4))) float v4f;
typedef __attribute__((ext_vector_type(8))) float v8f;

#define B_DIM 2
#define S_LEN 2048
#define D_DIM 1536
#define H_NUM 12
#define HD    128

// ---------------------------------------------------------------------------
// C[M,N] = A[M,K] @ W[N,K]^T + bias[N], fp32 via V_WMMA_F32_16X16X4_F32.
// Block = 256 threads = 8 waves arranged 4(M) x 2(N); wave tile = 32x64;
// block tile = 128x128. M=4096, N=K=1536 divide evenly.
//
// f32 16x16x4 fragment layout (ISA 7.12.2): for A (16x4) and B (4x16) alike,
// lane L holds elements (row = L%16, k = 2*(L/16) + {0,1}) -> one v2f load.
// C/D: lane L covers col = L%16; VGPR r covers row = r + 8*(L/16).
//
// Depth-2 software pipeline (P/Q fragment slots, k step 8): loads issued in
// iteration i are consumed in iteration i+1, so s_wait_loadcnt sits a full
// body (12 loads + 16 WMMAs) after the loads. 32-bit per-lane offsets off the
// uniform base pointer give saddr-form loads (no 64-bit VALU address chains).
// Working set is L2-resident (2x9.4MB weights + 25MB activations << 192MB).
// ---------------------------------------------------------------------------
__global__ __launch_bounds__(256)
void gemm_nt_bias_wmma(const float* __restrict__ A,
                       const float* __restrict__ W,
                       const float* __restrict__ bias,
                       float* __restrict__ C,
                       int M, int N, int K) {
  const int tilesM = M >> 7;
  const int bm = blockIdx.x % tilesM;
  const int bn = blockIdx.x / tilesM;

  const int tid   = threadIdx.x;
  const int lane  = tid & 31;
  const int wave  = tid >> 5;
  const int wm    = wave & 3;   // 0..3 -> 32-row strip
  const int wn    = wave >> 2;  // 0..1 -> 64-col strip
  const int lrow  = lane & 15;
  const int lhalf = lane >> 4;

  const int rowBase = (bm << 7) + (wm << 5);
  const int colBase = (bn << 7) + (wn << 6);

  // 32-bit per-lane element offsets from uniform base pointers
  const unsigned ao0 = (unsigned)(rowBase + lrow) * (unsigned)K + 2u * (unsigned)lhalf;
  const unsigned ao1 = ao0 + 16u * (unsigned)K;
  const unsigned wo0 = (unsigned)(colBase + lrow) * (unsigned)K + 2u * (unsigned)lhalf;
  const unsigned wo1 = wo0 + 16u * (unsigned)K;
  const unsigned wo2 = wo0 + 32u * (unsigned)K;
  const unsigned wo3 = wo0 + 48u * (unsigned)K;

#define LDA(off, kkk) (*(const v2f*)(A + (off) + (unsigned)(kkk)))
#define LDW(off, kkk) (*(const v2f*)(W + (off) + (unsigned)(kkk)))

#define WMMA8(A0, A1, B0, B1, B2, B3)                                                                   \
  do {                                                                                                  \
    acc[0][0] = __builtin_amdgcn_wmma_f32_16x16x4_f32(false, A0, false, B0, (short)0, acc[0][0], false, false); \
    acc[0][1] = __builtin_amdgcn_wmma_f32_16x16x4_f32(false, A0, false, B1, (short)0, acc[0][1], false, false); \
    acc[0][2] = __builtin_amdgcn_wmma_f32_16x16x4_f32(false, A0, false, B2, (short)0, acc[0][2], false, false); \
    acc[0][3] = __builtin_amdgcn_wmma_f32_16x16x4_f32(false, A0, false, B3, (short)0, acc[0][3], false, false); \
    acc[1][0] = __builtin_amdgcn_wmma_f32_16x16x4_f32(false, A1, false, B0, (short)0, acc[1][0], false, false); \
    acc[1][1] = __builtin_amdgcn_wmma_f32_16x16x4_f32(false, A1, false, B1, (short)0, acc[1][1], false, false); \
    acc[1][2] = __builtin_amdgcn_wmma_f32_16x16x4_f32(false, A1, false, B2, (short)0, acc[1][2], false, false); \
    acc[1][3] = __builtin_amdgcn_wmma_f32_16x16x4_f32(false, A1, false, B3, (short)0, acc[1][3], false, false); \
  } while (0)

  v8f acc[2][4] = {};

  // prologue: fill both pipeline slots (k = 0 and k = 4)
  v2f pa0 = LDA(ao0, 0), pa1 = LDA(ao1, 0);
  v2f pb0 = LDW(wo0, 0), pb1 = LDW(wo1, 0), pb2 = LDW(wo2, 0), pb3 = LDW(wo3, 0);
  v2f qa0 = LDA(ao0, 4), qa1 = LDA(ao1, 4);
  v2f qb0 = LDW(wo0, 4), qb1 = LDW(wo1, 4), qb2 = LDW(wo2, 4), qb3 = LDW(wo3, 4);

#pragma unroll 1
  for (int kk = 8; kk < K; kk += 8) {
    // consume P slot (data for kk-8), refill with kk
    WMMA8(pa0, pa1, pb0, pb1, pb2, pb3);
    pa0 = LDA(ao0, kk);
    pa1 = LDA(ao1, kk);
    pb0 = LDW(wo0, kk);
    pb1 = LDW(wo1, kk);
    pb2 = LDW(wo2, kk);
    pb3 = LDW(wo3, kk);
    __builtin_prefetch(A + ao0 + (unsigned)(kk + 64), 0, 3);

    // consume Q slot (data for kk-4), refill with kk+4
    WMMA8(qa0, qa1, qb0, qb1, qb2, qb3);
    qa0 = LDA(ao0, kk + 4);
    qa1 = LDA(ao1, kk + 4);
    qb0 = LDW(wo0, kk + 4);
    qb1 = LDW(wo1, kk + 4);
    qb2 = LDW(wo2, kk + 4);
    qb3 = LDW(wo3, kk + 4);
    __builtin_prefetch(W + wo0 + (unsigned)(kk + 64), 0, 3);
  }

  // epilogue: drain both slots (k = K-8, K-4)
  WMMA8(pa0, pa1, pb0, pb1, pb2, pb3);
  WMMA8(qa0, qa1, qb0, qb1, qb2, qb3);

#undef WMMA8
#undef LDA
#undef LDW

#pragma unroll
  for (int n = 0; n < 4; ++n) {
    const int col = colBase + n * 16 + lrow;
    const float bs = bias[col];
#pragma unroll
    for (int m = 0; m < 2; ++m) {
#pragma unroll
      for (int r = 0; r < 8; ++r) {
        const int row = rowBase + m * 16 + r + 8 * lhalf;
        C[(size_t)row * N + col] = acc[m][n][r] + bs;
      }
    }
  }
}

// ---------------------------------------------------------------------------
// Sliding-window uniform attention as a running sum per channel.
// Layouts are [B*S, D] row-major; channel = h*HD + d.
// Heads with small windows are chunked across S (warm-up <= window) so the
// grid stays reasonably parallel; big-window heads run serially (O(S) anyway).
// ---------------------------------------------------------------------------
__global__ __launch_bounds__(128)
void window_avg(const float* __restrict__ v, float* __restrict__ o) {
  const int bid = blockIdx.x;            // (b*H + h)*8 + chunk
  const int c = bid & 7;
  const int h = (bid >> 3) % H_NUM;
  const int b = bid / (8 * H_NUM);
  const int nch = (h <= 7) ? 8 : (h == 8 ? 4 : (h == 9 ? 2 : 1));
  if (c >= nch) return;

  const int w = (2 << h) - 1;            // 2^(h+1)-1
  const int chunk = S_LEN / nch;
  const int q0 = c * chunk, q1 = q0 + chunk;
  int start = q0 - w + 1;
  if (start < 0) start = 0;

  const int d = threadIdx.x;
  const size_t base = (size_t)b * S_LEN * D_DIM + (size_t)h * HD + d;

  float acc = 0.0f;
  for (int i = start; i < q1; ++i) {
    acc += v[base + (size_t)i * D_DIM];
    const int is = i - w;
    if (is >= start) acc -= v[base + (size_t)is * D_DIM];
    if (i >= q0) {
      const int len = (i + 1 < w) ? (i + 1) : w;
      o[base + (size_t)i * D_DIM] = acc * (1.0f / (float)len);
    }
  }
}

// ---------------------------------------------------------------------------
// attn_weights[b,h,i,j] = (i-w < j <= i) ? 1/min(i+1,w) : 0.
// 402 MB > 192 MB L2 -> non-temporal wide stores so it doesn't evict the
// L2-resident GEMM working set.
// ---------------------------------------------------------------------------
__global__ __launch_bounds__(512)
void attn_fill(float* __restrict__ attn) {
  const int bid = blockIdx.x;            // (b*H + h)*S + i
  const int i = bid & (S_LEN - 1);
  const int h = (bid / S_LEN) % H_NUM;
  const int w = (2 << h) - 1;
  const int len = (i + 1 < w) ? (i + 1) : w;
  const float inv = 1.0f / (float)len;
  const int lo = i - w;                  // j must satisfy lo < j <= i
  const int j0 = (int)threadIdx.x << 2;

  v4f val;
#pragma unroll
  for (int t = 0; t < 4; ++t) {
    const int j = j0 + t;
    val[t] = (j <= i && j > lo) ? inv : 0.0f;
  }
  __builtin_nontemporal_store(val, (v4f*)(attn + (size_t)bid * S_LEN + j0));
}

extern "C" void kernel_launch(void* const* d_in, const int* in_sizes, int n_in,
                              void* d_out, int out_size, void* d_ws, size_t ws_size,
                              hipStream_t stream) {
  const float* hidden = (const float*)d_in[0];
  const float* W_fc   = (const float*)d_in[1];
  const float* b_fc   = (const float*)d_in[2];
  const float* W_proj = (const float*)d_in[3];
  const float* b_proj = (const float*)d_in[4];

  float* out  = (float*)d_out;                                  // [B,S,D]
  float* attn = out + (size_t)B_DIM * S_LEN * D_DIM;            // [B,H,S,S]

  const int M = B_DIM * S_LEN;   // 4096
  const int N = D_DIM;           // 1536
  const int K = D_DIM;           // 1536

  // Stage v and the windowed output inside the attn region (overwritten last).
  float* vbuf = attn;                       // 25.2 MB
  float* obuf = attn + (size_t)M * N;       // next 25.2 MB (region is 402 MB)

  const int gemmBlocks = (M / 128) * (N / 128);   // 384

  // v = hidden @ W_fc^T + b_fc
  gemm_nt_bias_wmma<<<gemmBlocks, 256, 0, stream>>>(hidden, W_fc, b_fc, vbuf, M, N, K);
  // sliding-window uniform attention (per-head window), [B,H,S,hd] fused
  // directly back into [B,S,D] channel layout
  window_avg<<<B_DIM * H_NUM * 8, 128, 0, stream>>>(vbuf, obuf);
  // out = obuf @ W_proj^T + b_proj   (must run before attn region is clobbered)
  gemm_nt_bias_wmma<<<gemmBlocks, 256, 0, stream>>>(obuf, W_proj, b_proj, out, M, N, K);
  // materialize attn_weights last, overwriting the staging buffers
  attn_fill<<<B_DIM * H_NUM * S_LEN, 512, 0, stream>>>(attn);
}